// QuantumKernelNN_26242250179229
// MI455X (gfx1250) — compile-verified
//
#include <hip/hip_runtime.h>

typedef float v2f __attribute__((ext_vector_type(2)));
typedef float v8f __attribute__((ext_vector_type(8)));
typedef int   v4i_ __attribute__((vector_size(16)));   // matches builtin param type

#define AS1 __attribute__((address_space(1)))
#define AS3 __attribute__((address_space(3)))

#if defined(__has_builtin)
#if __has_builtin(__builtin_amdgcn_global_load_async_to_lds_b128) && \
    __has_builtin(__builtin_amdgcn_s_wait_asynccnt)
#define USE_ASYNC_LDS 1
#endif
#endif

#ifdef USE_ASYNC_LDS
#define ASYNC_G2L_B128(gsrc, ldst) \
    __builtin_amdgcn_global_load_async_to_lds_b128((AS1 v4i_*)(gsrc), (AS3 v4i_*)(ldst), 0, 0)
#endif

#define NSAMP 256
#define DIN   64
#define DOUT  120
#define QMODES 16
#define NPH   8

// ---------------------------------------------------------------------------
// Kernel 1: x_emb = sigmoid(x @ W^T + b) via V_WMMA_F32_16X16X4_F32
// Grid (16 Mtiles, 8 Ntiles), 1 wave per block. W row clamped (not guarded):
// C columns >= 120 compute garbage but are never stored -> no EXEC churn.
// ---------------------------------------------------------------------------
__global__ __launch_bounds__(32)
void k_embed(const float* __restrict__ x, const float* __restrict__ W,
             const float* __restrict__ b, float* __restrict__ xe) {
    const int lane  = threadIdx.x;
    const int mt    = blockIdx.x;       // row tile of x (16 rows)
    const int nt    = blockIdx.y;       // col tile (= W row tile)
    const int rsel  = lane & 15;
    const int khalf = lane >> 4;        // 0/1 -> K base offset 0/2
    const int row_a = mt * 16 + rsel;   // always < 256
    int row_w = nt * 16 + rsel;
    if (row_w >= DOUT) row_w = 0;       // clamp: keeps loads in-bounds only

    const float2* xp = (const float2*)(x + row_a * DIN);
    const float2* wp = (const float2*)(W + row_w * DIN);

    v8f acc = {0.f, 0.f, 0.f, 0.f, 0.f, 0.f, 0.f, 0.f};
#pragma unroll
    for (int kk = 0; kk < DIN; kk += 4) {
        const int h = (kk >> 1) + khalf;          // (kk + 2*khalf)/2
        const float2 a2 = xp[h];
        const float2 b2 = wp[h];
        const v2f af = {a2.x, a2.y};
        const v2f bf = {b2.x, b2.y};
        acc = __builtin_amdgcn_wmma_f32_16x16x4_f32(false, af, false, bf,
                                                    (short)0, acc, false, false);
    }
    // C layout: VGPR v -> lanes 0-15: M=v, N=lane ; lanes 16-31: M=v+8, N=lane-16
    const int col = nt * 16 + rsel;
    if (col < DOUT) {
        const float bias = b[col];
#pragma unroll
        for (int v = 0; v < 8; ++v) {
            const int row = mt * 16 + v + 8 * khalf;
            const float val = acc[v] + bias;
            xe[row * DOUT + col] = 1.f / (1.f + __expf(-val));
        }
    }
}

// ---------------------------------------------------------------------------
// Kernel 2: build per-sample unitary U (16x16 complex) from 60 2x2 rotations,
// one thread per (sample, U-column); column stays in registers.
// Writes V planes: VR[n][i][m], VI[n][i][m], VIn = -VI  (i<8 = output modes).
// ---------------------------------------------------------------------------
__global__ __launch_bounds__(256)
void k_unitary(const float* __restrict__ xe, float* __restrict__ VR,
               float* __restrict__ VI, float* __restrict__ VIn) {
    __shared__ float xs[16 * DOUT];
    const int tid = threadIdx.x;
    const int blk = blockIdx.x;                       // 16 blocks x 16 samples
    for (int idx = tid; idx < 16 * DOUT; idx += 256)
        xs[idx] = xe[blk * 16 * DOUT + idx];
    __syncthreads();

    const int nl = tid >> 4;          // local sample
    const int c  = tid & 15;          // column of U owned by this thread

    float ur[QMODES], ui[QMODES];
#pragma unroll
    for (int m = 0; m < QMODES; ++m) { ur[m] = (m == c) ? 1.f : 0.f; ui[m] = 0.f; }

    int nb_before = 0;
#pragma unroll
    for (int d = 0; d < 8; ++d) {
        const int off = d & 1;
        const int nb  = off ? 7 : 8;
#pragma unroll
        for (int k = 0; k < nb; ++k) {
            const int bi = nb_before + k;
            const float th = xs[nl * DOUT + 2 * bi]     * 1.57079632679489662f;
            const float ph = xs[nl * DOUT + 2 * bi + 1] * 6.28318530717958648f;
            const float ct = __cosf(th), st = __sinf(th);
            const float cp = __cosf(ph), sp = __sinf(ph);
            // block = [[e^{i phi} ct, -st], [e^{i phi} st, ct]]
            const float aRe = cp * ct, aIm = sp * ct;
            const float cRe = cp * st, cIm = sp * st;
            const int r0 = off + 2 * k, r1 = r0 + 1;
            const float u0r = ur[r0], u0i = ui[r0];
            const float u1r = ur[r1], u1i = ui[r1];
            ur[r0] = aRe * u0r - aIm * u0i - st * u1r;
            ui[r0] = aRe * u0i + aIm * u0r - st * u1i;
            ur[r1] = cRe * u0r - cIm * u0i + ct * u1r;
            ui[r1] = cRe * u0i + cIm * u0r + ct * u1i;
        }
        nb_before += nb;
    }
    if (c < NPH) {
        const int n = blk * 16 + nl;
        const int base = (n * NPH + c) * QMODES;
#pragma unroll
        for (int m = 0; m < QMODES; ++m) {
            VR[base + m]  = ur[m];
            VI[base + m]  = ui[m];
            VIn[base + m] = -ui[m];
        }
    }
}

// ---------------------------------------------------------------------------
// Kernel 3: per 8x8 (a,b) tile: A[a,b] = V[a]^H V[b] via WMMA (64x64x16 real
// GEMMs), then Ryser permanent (Gray code) per thread. K = |perm|^2, diag=1.
// ---------------------------------------------------------------------------
__global__ __launch_bounds__(64)
void k_perm(const float* __restrict__ VR, const float* __restrict__ VI,
            const float* __restrict__ VIn, float* __restrict__ K) {
    __shared__ __align__(16) float PAR[64 * 16], PAI[64 * 16], PAIn[64 * 16];
    __shared__ __align__(16) float PBR[64 * 16], PBI[64 * 16];
    __shared__ float Are[64 * 66], Aim[64 * 66];   // stride 66 -> conflict-free

    const int tid = threadIdx.x;
    const int a0 = blockIdx.y * 8, b0 = blockIdx.x * 8;

    // Phase 1: stage V planes.  Row r=(a_local*8+i) -> global offset a0*128+idx.
#ifdef USE_ASYNC_LDS
    for (int idx = tid * 4; idx < 64 * 16; idx += 64 * 4) {
        ASYNC_G2L_B128(VR  + a0 * 128 + idx, &PAR[idx]);
        ASYNC_G2L_B128(VI  + a0 * 128 + idx, &PAI[idx]);
        ASYNC_G2L_B128(VIn + a0 * 128 + idx, &PAIn[idx]);
        ASYNC_G2L_B128(VR  + b0 * 128 + idx, &PBR[idx]);
        ASYNC_G2L_B128(VI  + b0 * 128 + idx, &PBI[idx]);
    }
    __builtin_amdgcn_s_wait_asynccnt(0);
#else
    for (int idx = tid * 4; idx < 64 * 16; idx += 64 * 4) {
        *(float4*)&PAR[idx]  = *(const float4*)(VR  + a0 * 128 + idx);
        *(float4*)&PAI[idx]  = *(const float4*)(VI  + a0 * 128 + idx);
        *(float4*)&PAIn[idx] = *(const float4*)(VIn + a0 * 128 + idx);
        *(float4*)&PBR[idx]  = *(const float4*)(VR  + b0 * 128 + idx);
        *(float4*)&PBI[idx]  = *(const float4*)(VI  + b0 * 128 + idx);
    }
#endif
    __syncthreads();

    // Phase 2: 4x4 grid of 16x16 WMMA C tiles over M=64 (a,i), N=64 (b,j), K=16.
    const int wave  = tid >> 5;
    const int lane  = tid & 31;
    const int rsel  = lane & 15;
    const int khalf = lane >> 4;
#pragma unroll
    for (int t = wave; t < 16; t += 2) {
        const int tm = t >> 2, tn = t & 3;
        const int arow = (tm * 16 + rsel) * 16;   // A-side plane row base
        const int brow = (tn * 16 + rsel) * 16;   // B-side plane row base
        v8f cre = {0.f,0.f,0.f,0.f,0.f,0.f,0.f,0.f};
        v8f cim = {0.f,0.f,0.f,0.f,0.f,0.f,0.f,0.f};
#pragma unroll
        for (int kb = 0; kb < 16; kb += 4) {
            const int h = (kb >> 1) + khalf;      // (kb + 2*khalf)/2
            const float2 aR = ((const float2*)(PAR  + arow))[h];
            const float2 aI = ((const float2*)(PAI  + arow))[h];
            const float2 aN = ((const float2*)(PAIn + arow))[h];
            const float2 bR = ((const float2*)(PBR  + brow))[h];
            const float2 bI = ((const float2*)(PBI  + brow))[h];
            const v2f vaR = {aR.x, aR.y}, vaI = {aI.x, aI.y}, vaN = {aN.x, aN.y};
            const v2f vbR = {bR.x, bR.y}, vbI = {bI.x, bI.y};
            // Re(A) = VaR.VbR + VaI.VbI ; Im(A) = VaR.VbI + (-VaI).VbR
            cre = __builtin_amdgcn_wmma_f32_16x16x4_f32(false, vaR, false, vbR, (short)0, cre, false, false);
            cre = __builtin_amdgcn_wmma_f32_16x16x4_f32(false, vaI, false, vbI, (short)0, cre, false, false);
            cim = __builtin_amdgcn_wmma_f32_16x16x4_f32(false, vaR, false, vbI, (short)0, cim, false, false);
            cim = __builtin_amdgcn_wmma_f32_16x16x4_f32(false, vaN, false, vbR, (short)0, cim, false, false);
        }
#pragma unroll
        for (int v = 0; v < 8; ++v) {
            const int rM = tm * 16 + v + 8 * khalf;   // (a_local, i)
            const int cN = tn * 16 + rsel;            // (b_local, j)
            const int p  = (rM >> 3) * 8 + (cN >> 3);
            const int i  = rM & 7, j = cN & 7;
            Are[p * 66 + j * 8 + i] = cre[v];   // column-major per pair
            Aim[p * 66 + j * 8 + i] = cim[v];
        }
    }
    __syncthreads();

    // Phase 3: Ryser permanent via Gray code, one (a,b) pair per thread.
    // Odd steps (t = 2m+1) always toggle column 0 (register-cached, dir = parity of m).
    const int p = tid;
    const float* ar = &Are[p * 66];
    const float* ai = &Aim[p * 66];
    float rsr[8], rsi[8], c0r[8], c0i[8];
#pragma unroll
    for (int i = 0; i < 8; ++i) {
        rsr[i] = 0.f; rsi[i] = 0.f;
        c0r[i] = ar[i]; c0i[i] = ai[i];
    }
    float accr = 0.f, acci = 0.f;
    for (int m = 0; m < 128; ++m) {
        // ---- t = 2m+1 (odd): tz==0, dir alternates, sign = -1
        {
            const float dir = (m & 1) ? -1.f : 1.f;
#pragma unroll
            for (int i = 0; i < 8; ++i) {
                rsr[i] = fmaf(dir, c0r[i], rsr[i]);
                rsi[i] = fmaf(dir, c0i[i], rsi[i]);
            }
            float pr = rsr[0], pi = rsi[0];
#pragma unroll
            for (int i = 1; i < 8; ++i) {
                const float nr = pr * rsr[i] - pi * rsi[i];
                const float ni = pr * rsi[i] + pi * rsr[i];
                pr = nr; pi = ni;
            }
            accr -= pr; acci -= pi;                   // sgn = -1 for odd t
        }
        // ---- t = 2m+2 (even): column tz from LDS, sign = +1
        const int t = 2 * m + 2;
        if (t < 256) {
            const int tz = __builtin_ctz(t);
            const int g  = t ^ (t >> 1);
            const float dir = ((g >> tz) & 1) ? 1.f : -1.f;
            const float* cr = &ar[tz * 8];
            const float* ci = &ai[tz * 8];
#pragma unroll
            for (int i = 0; i < 8; ++i) {
                rsr[i] = fmaf(dir, cr[i], rsr[i]);
                rsi[i] = fmaf(dir, ci[i], rsi[i]);
            }
            float pr = rsr[0], pi = rsi[0];
#pragma unroll
            for (int i = 1; i < 8; ++i) {
                const float nr = pr * rsr[i] - pi * rsi[i];
                const float ni = pr * rsi[i] + pi * rsr[i];
                pr = nr; pi = ni;
            }
            accr += pr; acci += pi;                   // sgn = +1 for even t
        }
    }
    const int a = a0 + (p >> 3), bb = b0 + (p & 7);
    K[a * NSAMP + bb] = (a == bb) ? 1.f : (accr * accr + acci * acci);
}

// ---------------------------------------------------------------------------
extern "C" void kernel_launch(void* const* d_in, const int* in_sizes, int n_in,
                              void* d_out, int out_size, void* d_ws, size_t ws_size,
                              hipStream_t stream) {
    (void)in_sizes; (void)n_in; (void)out_size; (void)ws_size;
    const float* x = (const float*)d_in[0];
    const float* W = (const float*)d_in[1];
    const float* b = (const float*)d_in[2];
    float* out = (float*)d_out;
    float* xe  = out;                      // (256,120) first output
    float* K   = out + NSAMP * DOUT;       // (256,256) second output
    float* VR  = (float*)d_ws;             // 256*8*16 floats each
    float* VI  = VR + NSAMP * NPH * QMODES;
    float* VIn = VI + NSAMP * NPH * QMODES;

    k_embed  <<<dim3(16, 8),  32,  0, stream>>>(x, W, b, xe);
    k_unitary<<<16,           256, 0, stream>>>(xe, VR, VI, VIn);
    k_perm   <<<dim3(32, 32), 64,  0, stream>>>(VR, VI, VIn, K);
}